// FullAttention_45054206935095
// MI455X (gfx1250) — compile-verified
//
#include <hip/hip_runtime.h>
#include <math.h>

// LogSparse + causal attention, B=4 L=2048 H=8 E=D=64, fp32.
// Allowed relative offsets d = q - s: {0..7, 9, 13, 21, 37, 69, 133, 261, 517, 1029}
// (<=17 keys per query). Flash-attention over only the active 16x16 tiles,
// fp32 WMMA (v_wmma_f32_16x16x4_f32) for both Q.K^T and P.V.
// Softmax row reductions done via LDS row-major re-read (2 b128 loads) +
// a single cross-half shuffle, instead of 8x4 ds_bpermute butterflies.

typedef float v2f __attribute__((ext_vector_type(2)));
typedef float v8f __attribute__((ext_vector_type(8)));

#define WMMA_F32X4(a, b, c) \
  __builtin_amdgcn_wmma_f32_16x16x4_f32(false, (a), false, (b), (short)0, (c), false, false)

__launch_bounds__(32)
__global__ void logsparse_attn_kernel(const float* __restrict__ Q,
                                      const float* __restrict__ K,
                                      const float* __restrict__ V,
                                      float* __restrict__ O)
{
    constexpr int B = 4, L = 2048, H = 8, E = 64, D = 64;
    constexpr int NQT = L / 16;                 // 128 query tiles per (b,h)
    const size_t rowQ = (size_t)H * E;          // 512 floats between consecutive l
    const size_t rowV = (size_t)H * D;          // 512 floats

    const int lane = threadIdx.x;               // wave32
    const int n16  = lane & 15;
    const int half = lane >> 4;                 // 0 or 1

    int bid = blockIdx.x;
    const int qt = bid & (NQT - 1);
    bid >>= 7;
    const int hh = bid & (H - 1);
    const int bb = bid >> 3;
    const int q0 = qt * 16;

    // ---- Load Q tile into WMMA-A f32 16x4 layout, pre-scaled by 1/sqrt(E) ----
    const float* Qbase = Q + ((size_t)((size_t)bb * L + q0) * H + hh) * E;
    v2f aq[16];
#pragma unroll
    for (int c = 0; c < 16; ++c) {
        const int k0 = 4 * c + 2 * half;
        const float2 qv = *(const float2*)(Qbase + (size_t)n16 * rowQ + k0);
        aq[c].x = qv.x * 0.125f;
        aq[c].y = qv.y * 0.125f;
    }

    // Output accumulators: 4 N-tiles of 16 dims, C/D layout (lane=n, vgpr r -> row r+8*half)
    v8f o0 = {}, o1 = {}, o2 = {}, o3 = {};
    // Row softmax stats: this lane owns row (lane%16); duplicated across halves.
    float mrow = -INFINITY, lrow = 0.0f;

    // ---- Active key-tile bitmask (uniform across the wave) ----
    const int DV[17] = {0,1,2,3,4,5,6,7,9,13,21,37,69,133,261,517,1029};
    unsigned long long bm0 = 0, bm1 = 0;
#pragma unroll
    for (int i = 0; i < 17; ++i) {
        const int d = DV[i];
        int shi = q0 + 15 - d; if (shi < 0) continue;
        int slo = q0 - d;      if (slo < 0) slo = 0;
        const int t0 = slo >> 4, t1 = shi >> 4;
        if (t0 < 64) bm0 |= 1ull << t0; else bm1 |= 1ull << (t0 - 64);
        if (t1 < 64) bm0 |= 1ull << t1; else bm1 |= 1ull << (t1 - 64);
    }

    const unsigned long long SMALL =
        0xFFull | (1ull << 9) | (1ull << 13) | (1ull << 21) | (1ull << 37);

    __shared__ __align__(16) float sP[16 * 16];  // scores -> P staging
    __shared__ __align__(16) float sA[16];       // per-row alpha broadcast
    __shared__ __align__(16) float sL[16];       // per-row 1/l broadcast

    const int rowoff = n16 * 16 + 8 * half;      // this lane's row-major half-row

    for (int t = 0; t <= qt; ++t) {
        const bool act = (t < 64) ? ((bm0 >> t) & 1ull) : ((bm1 >> (t - 64)) & 1ull);
        if (!act) continue;
        const int s0 = 16 * t;

        // ---- S = (Q*scale) @ K^T: 16 chained f32 WMMAs ----
        const float* Kbase = K + ((size_t)((size_t)bb * L + s0) * H + hh) * E;
        v8f s = {};
#pragma unroll
        for (int c = 0; c < 16; ++c) {
            const int k0 = 4 * c + 2 * half;
            const float2 kv = *(const float2*)(Kbase + (size_t)n16 * rowQ + k0);
            v2f bk; bk.x = kv.x; bk.y = kv.y;
            s = WMMA_F32X4(aq[c], bk, s);
        }

        // ---- masked scores to LDS (C layout) ----
#pragma unroll
        for (int r = 0; r < 8; ++r) {
            const int row = r + 8 * half;
            const int d = (q0 + row) - (s0 + n16);
            const bool ok = (d >= 0) &&
                ((d < 64) ? (((SMALL >> d) & 1ull) != 0)
                          : (d == 69 || d == 133 || d == 261 || d == 517 || d == 1029));
            sP[row * 16 + n16] = ok ? s[r] : -INFINITY;
        }
        __syncthreads();

        // ---- row-major softmax: lane owns row n16, cols [8*half, 8*half+8) ----
        float4 p0 = *(const float4*)&sP[rowoff];
        float4 p1 = *(const float4*)&sP[rowoff + 4];
        float rmax = fmaxf(fmaxf(fmaxf(p0.x, p0.y), fmaxf(p0.z, p0.w)),
                           fmaxf(fmaxf(p1.x, p1.y), fmaxf(p1.z, p1.w)));
        rmax = fmaxf(rmax, __shfl_xor(rmax, 16, 32));      // combine halves
        const float mnew  = fmaxf(mrow, rmax);
        const float msafe = (mnew == -INFINITY) ? 0.0f : mnew;  // -inf-safe
        const float alpha = __expf(mrow - msafe);               // exp(-inf)=0
        mrow = mnew;
        p0.x = __expf(p0.x - msafe); p0.y = __expf(p0.y - msafe);
        p0.z = __expf(p0.z - msafe); p0.w = __expf(p0.w - msafe);
        p1.x = __expf(p1.x - msafe); p1.y = __expf(p1.y - msafe);
        p1.z = __expf(p1.z - msafe); p1.w = __expf(p1.w - msafe);
        float rs = (p0.x + p0.y) + (p0.z + p0.w) + (p1.x + p1.y) + (p1.z + p1.w);
        rs += __shfl_xor(rs, 16, 32);                       // full row sum
        lrow = lrow * alpha + rs;
        // write P back (same addresses this lane just read: no cross-lane hazard)
        *(float4*)&sP[rowoff]     = p0;
        *(float4*)&sP[rowoff + 4] = p1;
        if (half == 0) sA[n16] = alpha;                     // per-row alpha
        __syncthreads();

        // ---- rescale running output by per-row alpha (C layout) ----
        const float4 a0 = *(const float4*)&sA[8 * half];
        const float4 a1 = *(const float4*)&sA[8 * half + 4];
        const float ar[8] = {a0.x, a0.y, a0.z, a0.w, a1.x, a1.y, a1.z, a1.w};
#pragma unroll
        for (int r = 0; r < 8; ++r) {
            o0[r] *= ar[r]; o1[r] *= ar[r]; o2[r] *= ar[r]; o3[r] *= ar[r];
        }

        // ---- P in A layout from LDS (8B-aligned b64 reads) ----
        v2f pa[4];
#pragma unroll
        for (int c = 0; c < 4; ++c) {
            const int k0 = 4 * c + 2 * half;
            pa[c] = *(const v2f*)&sP[n16 * 16 + k0];
        }

        // ---- O += P @ V_tile: 16 chained f32 WMMAs over 4 N-tiles ----
        const float* Vbase = V + ((size_t)((size_t)bb * L + s0) * H + hh) * D;
#pragma unroll
        for (int c = 0; c < 4; ++c) {
            const int k0 = 4 * c + 2 * half;
            const float* vr0 = Vbase + (size_t)k0 * rowV;
            const float* vr1 = vr0 + rowV;
            v2f bv0, bv1, bv2, bv3;
            bv0.x = vr0[ 0 + n16]; bv0.y = vr1[ 0 + n16];
            bv1.x = vr0[16 + n16]; bv1.y = vr1[16 + n16];
            bv2.x = vr0[32 + n16]; bv2.y = vr1[32 + n16];
            bv3.x = vr0[48 + n16]; bv3.y = vr1[48 + n16];
            o0 = WMMA_F32X4(pa[c], bv0, o0);
            o1 = WMMA_F32X4(pa[c], bv1, o1);
            o2 = WMMA_F32X4(pa[c], bv2, o2);
            o3 = WMMA_F32X4(pa[c], bv3, o3);
        }
        __syncthreads();   // protect sP/sA before next tile overwrites them
    }

    // ---- broadcast 1/l per row, normalize and store (coalesced) ----
    if (half == 0) sL[n16] = 1.0f / lrow;   // lrow > 0: diagonal key always allowed
    __syncthreads();
    const float4 l0 = *(const float4*)&sL[8 * half];
    const float4 l1 = *(const float4*)&sL[8 * half + 4];
    const float lr[8] = {l0.x, l0.y, l0.z, l0.w, l1.x, l1.y, l1.z, l1.w};

    float* Obase = O + ((size_t)((size_t)bb * L + q0) * H + hh) * D;
#pragma unroll
    for (int r = 0; r < 8; ++r) {
        const int row = r + 8 * half;
        float* orow = Obase + (size_t)row * rowV;
        orow[ 0 + n16] = o0[r] * lr[r];
        orow[16 + n16] = o1[r] * lr[r];
        orow[32 + n16] = o2[r] * lr[r];
        orow[48 + n16] = o3[r] * lr[r];
    }
}

extern "C" void kernel_launch(void* const* d_in, const int* in_sizes, int n_in,
                              void* d_out, int out_size, void* d_ws, size_t ws_size,
                              hipStream_t stream) {
    (void)in_sizes; (void)n_in; (void)out_size; (void)d_ws; (void)ws_size;
    const float* Q = (const float*)d_in[0];
    const float* K = (const float*)d_in[1];
    const float* V = (const float*)d_in[2];
    float* out = (float*)d_out;
    // One wave32 per (b, h, 16-query tile): 4*8*128 = 4096 workgroups
    dim3 grid(4 * 8 * 128), block(32);
    logsparse_attn_kernel<<<grid, block, 0, stream>>>(Q, K, V, out);
}